// EquivariantLayer_40742059770228
// MI455X (gfx1250) — compile-verified
//
#include <hip/hip_runtime.h>

typedef __attribute__((ext_vector_type(2))) float v2f;
typedef __attribute__((ext_vector_type(8))) float v8f;

#define BATCH 8
#define NROWS 65536
#define CDIM  64
#define PART_PER_BATCH 64            // partial-sum blocks per batch
#define ROWS_PER_PART (NROWS / PART_PER_BATCH)   // 1024
#define LDA_T 68                     // padded LDS stride (conflict-free b64 reads)

// ---------------- Kernel 1: partial column sums (deterministic order) ----------------
__global__ __launch_bounds__(256) void colsum_partial(const float* __restrict__ x,
                                                      float* __restrict__ partial) {
    const int b  = blockIdx.x / PART_PER_BATCH;
    const int j  = blockIdx.x % PART_PER_BATCH;
    const int c  = threadIdx.x & 63;   // channel
    const int rg = threadIdx.x >> 6;   // row group 0..3
    const float* p = x + (((size_t)b * NROWS + (size_t)j * ROWS_PER_PART
                           + (size_t)rg * 256) * CDIM) + c;
    float s = 0.f;
#pragma unroll 8
    for (int r = 0; r < 256; ++r) s += p[(size_t)r * CDIM];  // coalesced 256B rows
    __shared__ float red[256];
    red[threadIdx.x] = s;
    __syncthreads();
    if (threadIdx.x < 64) {
        partial[(size_t)blockIdx.x * 64 + c] =
            red[c] + red[64 + c] + red[128 + c] + red[192 + c];   // fixed order
    }
}

// ---------------- Kernel 2: finish reduction, t[b] = s[b] @ Bm ----------------
__global__ __launch_bounds__(64) void finish_t(const float* __restrict__ partial,
                                               const float* __restrict__ Bm,
                                               float* __restrict__ t) {
    const int b = blockIdx.x;
    const int d = threadIdx.x;           // 0..63
    __shared__ float s[64];
    float acc = 0.f;
    for (int j = 0; j < PART_PER_BATCH; ++j)
        acc += partial[((size_t)b * PART_PER_BATCH + j) * 64 + d];  // fixed order
    s[d] = acc;
    __syncthreads();
    float tv = 0.f;
#pragma unroll
    for (int c = 0; c < 64; ++c) tv += s[c] * Bm[c * 64 + d];
    t[b * 64 + d] = tv;
}

// ---------------- Kernel 3: out = x @ A - t[b], via V_WMMA_F32_16X16X4_F32 ----------------
// Each wave: one 16-row strip, all 64 output columns (4 accumulator tiles).
__global__ __launch_bounds__(256) void gemm_wmma(const float* __restrict__ x,
                                                 const float* __restrict__ A,
                                                 const float* __restrict__ t,
                                                 float* __restrict__ out) {
    __shared__ float At[64 * LDA_T];   // A transposed: At[d*LDA_T + c] = A[c*64+d]
    for (int i = threadIdx.x; i < 4096; i += 256) {
        const int c = i >> 6, d = i & 63;
        At[d * LDA_T + c] = A[i];      // coalesced global read
    }
    __syncthreads();

    const int wave = threadIdx.x >> 5;
    const int lane = threadIdx.x & 31;
    const int lh   = lane >> 4;        // lane half (0/1)
    const int lm   = lane & 15;

    const int    strip = blockIdx.x * 8 + wave;
    const size_t row0  = (size_t)strip * 16;
    const int    b     = (int)(row0 >> 16);   // NROWS = 65536 rows per batch

    // A-operand (x) fragment source: lane lm holds row (row0+lm);
    // lanes 0-15 get K = k,k+1 (VGPR0,VGPR1); lanes 16-31 get K = k+2,k+3.
    const float* xa = x + (row0 + (size_t)lm) * CDIM + 2 * lh;

    v8f acc0 = {}, acc1 = {}, acc2 = {}, acc3 = {};
#pragma unroll
    for (int k = 0; k < CDIM; k += 4) {
        v2f af = *(const v2f*)(xa + k);                       // global_load_b64
        // B-operand (A matrix) fragments: column N = n0+lm, rows K = k+2*lh, k+1+2*lh
        const float* bb = &At[(size_t)lm * LDA_T + k + 2 * lh];
        v2f b0 = *(const v2f*)(bb);                           // ds_load_b64, bank-conflict free
        v2f b1 = *(const v2f*)(bb + 16 * LDA_T);
        v2f b2 = *(const v2f*)(bb + 32 * LDA_T);
        v2f b3 = *(const v2f*)(bb + 48 * LDA_T);
        acc0 = __builtin_amdgcn_wmma_f32_16x16x4_f32(false, af, false, b0, (short)0, acc0, false, false);
        acc1 = __builtin_amdgcn_wmma_f32_16x16x4_f32(false, af, false, b1, (short)0, acc1, false, false);
        acc2 = __builtin_amdgcn_wmma_f32_16x16x4_f32(false, af, false, b2, (short)0, acc2, false, false);
        acc3 = __builtin_amdgcn_wmma_f32_16x16x4_f32(false, af, false, b3, (short)0, acc3, false, false);
    }

    // Epilogue: subtract broadcast t[b] and store.
    const float* tb = t + b * 64 + lm;
    const float t0 = tb[0], t1 = tb[16], t2 = tb[32], t3 = tb[48];
    // C/D layout: VGPR i, lanes 0-15 -> row M=i; lanes 16-31 -> row M=i+8; col N=lane%16.
    float* o = out + (row0 + (size_t)lh * 8) * CDIM + lm;
#pragma unroll
    for (int i = 0; i < 8; ++i) {
        const size_t off = (size_t)i * CDIM;
        o[off +  0] = acc0[i] - t0;
        o[off + 16] = acc1[i] - t1;
        o[off + 32] = acc2[i] - t2;
        o[off + 48] = acc3[i] - t3;
    }
}

extern "C" void kernel_launch(void* const* d_in, const int* in_sizes, int n_in,
                              void* d_out, int out_size, void* d_ws, size_t ws_size,
                              hipStream_t stream) {
    const float* x  = (const float*)d_in[0];
    const float* A  = (const float*)d_in[1];
    const float* Bm = (const float*)d_in[2];
    float* out = (float*)d_out;

    float* partial = (float*)d_ws;                       // 512*64 floats = 128 KB
    float* t       = partial + BATCH * PART_PER_BATCH * 64;  // 512 floats

    colsum_partial<<<BATCH * PART_PER_BATCH, 256, 0, stream>>>(x, partial);
    finish_t<<<BATCH, 64, 0, stream>>>(partial, Bm, t);

    const int strips = (BATCH * NROWS) / 16;             // 32768 waves
    gemm_wmma<<<strips / 8, 256, 0, stream>>>(x, A, t, out);
}